// DiffusionGraphConv_34583076668041
// MI455X (gfx1250) — compile-verified
//
#include <hip/hip_runtime.h>

#define N_NODES 10000
#define P_DIM   32
#define Q_DIM   64
#define B_SZ    32
#define E_EDGES 160000
#define M_MATS  5
#define F_COLS  1024                      // B*P, inner layout b*32+p
#define XSZ     (N_NODES * F_COLS)        // floats per diffusion buffer
#define AS_STRIDE 164                     // padded LDS row stride (floats)

typedef __attribute__((ext_vector_type(2))) float v2f;
typedef __attribute__((ext_vector_type(8))) float v8f;

// ---------------- Kernel 1: inputs [B, N*P] -> x0 [N][b*32+p] ----------------
__global__ void __launch_bounds__(256)
k_build_x0(const float* __restrict__ in, float* __restrict__ x0) {
    int tid = blockIdx.x * 256 + threadIdx.x;   // n*1024 + b*32 + p
    int p = tid & 31;
    int b = (tid >> 5) & 31;
    int n = tid >> 10;
    x0[tid] = in[b * (N_NODES * P_DIM) + n * P_DIM + p];
}

// ---------------- Kernel 2: y = alpha * x0 (init for spmm accumulation) -----
__global__ void __launch_bounds__(256)
k_init(const float* __restrict__ x0, float* __restrict__ y, float alpha) {
    int tid = blockIdx.x * 256 + threadIdx.x;
    float4 v = ((const float4*)x0)[tid];
    float4 o = make_float4(alpha * v.x, alpha * v.y, alpha * v.z, alpha * v.w);
    ((float4*)y)[tid] = o;
}

// ---------------- Kernel 3: y += scale * A x  (one block per edge) ----------
__global__ void __launch_bounds__(256)
k_spmm(const int* __restrict__ rows, const int* __restrict__ cols,
       const float* __restrict__ vals, const float* __restrict__ x,
       float* __restrict__ y, float scale) {
    int e = blockIdx.x;
    int r = rows[e];
    int c = cols[e];
    float v = vals[e] * scale;
    int off = threadIdx.x * 4;
    const float4 xv = *(const float4*)(x + (size_t)c * F_COLS + off);
    float* yp = y + (size_t)r * F_COLS + off;
    unsafeAtomicAdd(yp + 0, v * xv.x);
    unsafeAtomicAdd(yp + 1, v * xv.y);
    unsafeAtomicAdd(yp + 2, v * xv.z);
    unsafeAtomicAdd(yp + 3, v * xv.w);
}

// ---------------- Kernel 4: fused gather + GEMM via V_WMMA_F32_16X16X4_F32 --
// Block: 64 threads (2 waves). Block computes 64 rows x 64 cols of
// out[320000, 64] = A[320000, 160] @ W[160, 64] + bias, where
// A[(b*N+n)][p*5+m] = xs[m][n*1024 + b*32 + p].
// Each wave handles TWO 16-row tiles so every loaded B-fragment feeds 2 WMMAs.
__global__ void __launch_bounds__(64)
k_gemm_wmma(const float* __restrict__ xs, const float* __restrict__ W,
            const float* __restrict__ bias, float* __restrict__ out) {
    __shared__ float As[64 * AS_STRIDE];

    const int t  = threadIdx.x;         // 0..63
    const int r0 = blockIdx.x * 64;

    // Cooperative gather of the 64x160 A-tile: 2560 float4 chunks, 40/thread.
    #pragma unroll 4
    for (int i = 0; i < 40; ++i) {
        int cid = t + i * 64;           // [0, 2560)
        int row = cid / 40;             // 0..63
        int rem = cid - row * 40;
        int m   = rem >> 3;             // 0..4  (diffusion matrix)
        int p4  = rem & 7;              // 0..7  (float4 within 32 p's)
        int r   = r0 + row;
        int b   = r / N_NODES;
        int n   = r - b * N_NODES;
        const float4 v = *(const float4*)(xs + (size_t)m * XSZ +
                                          (size_t)n * F_COLS + b * 32 + p4 * 4);
        float* dst = &As[row * AS_STRIDE];
        int p = p4 * 4;
        dst[(p + 0) * 5 + m] = v.x;
        dst[(p + 1) * 5 + m] = v.y;
        dst[(p + 2) * 5 + m] = v.z;
        dst[(p + 3) * 5 + m] = v.w;
    }
    __syncthreads();

    const int w    = t >> 5;            // wave id 0..1 -> 32-row subtile
    const int lane = t & 31;
    const int half = lane >> 4;         // 0: K=0,1   1: K=2,3
    const int lr   = lane & 15;

    const float* arowA = &As[(w * 32 + lr) * AS_STRIDE];        // rows +0..15
    const float* arowB = arowA + 16 * AS_STRIDE;                // rows +16..31

    float b0 = bias[ 0 + lr];
    float b1 = bias[16 + lr];
    float b2 = bias[32 + lr];
    float b3 = bias[48 + lr];
    v8f accA0 = {b0, b0, b0, b0, b0, b0, b0, b0};
    v8f accA1 = {b1, b1, b1, b1, b1, b1, b1, b1};
    v8f accA2 = {b2, b2, b2, b2, b2, b2, b2, b2};
    v8f accA3 = {b3, b3, b3, b3, b3, b3, b3, b3};
    v8f accB0 = accA0;
    v8f accB1 = accA1;
    v8f accB2 = accA2;
    v8f accB3 = accA3;

    #pragma unroll 4
    for (int kk = 0; kk < 40; ++kk) {
        int k0 = kk * 4 + half * 2;
        // A frags (16x4): lanes 0-15 hold M=lr with K=k0,k0+1; lanes 16-31 K+2
        v2f aA; aA.x = arowA[k0]; aA.y = arowA[k0 + 1];
        v2f aB; aB.x = arowB[k0]; aB.y = arowB[k0 + 1];
        // B frags (4x16): lane lr holds column q0+lr, K striped like A
        const float* wp = W + k0 * Q_DIM + lr;
        v2f bv0; bv0.x = wp[ 0]; bv0.y = wp[ 0 + Q_DIM];
        v2f bv1; bv1.x = wp[16]; bv1.y = wp[16 + Q_DIM];
        v2f bv2; bv2.x = wp[32]; bv2.y = wp[32 + Q_DIM];
        v2f bv3; bv3.x = wp[48]; bv3.y = wp[48 + Q_DIM];
        accA0 = __builtin_amdgcn_wmma_f32_16x16x4_f32(false, aA, false, bv0,
                                                      (short)0, accA0, false, false);
        accA1 = __builtin_amdgcn_wmma_f32_16x16x4_f32(false, aA, false, bv1,
                                                      (short)0, accA1, false, false);
        accA2 = __builtin_amdgcn_wmma_f32_16x16x4_f32(false, aA, false, bv2,
                                                      (short)0, accA2, false, false);
        accA3 = __builtin_amdgcn_wmma_f32_16x16x4_f32(false, aA, false, bv3,
                                                      (short)0, accA3, false, false);
        accB0 = __builtin_amdgcn_wmma_f32_16x16x4_f32(false, aB, false, bv0,
                                                      (short)0, accB0, false, false);
        accB1 = __builtin_amdgcn_wmma_f32_16x16x4_f32(false, aB, false, bv1,
                                                      (short)0, accB1, false, false);
        accB2 = __builtin_amdgcn_wmma_f32_16x16x4_f32(false, aB, false, bv2,
                                                      (short)0, accB2, false, false);
        accB3 = __builtin_amdgcn_wmma_f32_16x16x4_f32(false, aB, false, bv3,
                                                      (short)0, accB3, false, false);
    }

    // D layout: VGPR j, lanes 0-15 -> row j, col lr; lanes 16-31 -> row j+8
    #pragma unroll
    for (int j = 0; j < 8; ++j) {
        int rgA = r0 + w * 32 + j + half * 8;
        float* opA = out + (size_t)rgA * Q_DIM + lr;
        opA[ 0] = accA0[j];
        opA[16] = accA1[j];
        opA[32] = accA2[j];
        opA[48] = accA3[j];
        float* opB = opA + (size_t)16 * Q_DIM;
        opB[ 0] = accB0[j];
        opB[16] = accB1[j];
        opB[32] = accB2[j];
        opB[48] = accB3[j];
    }
}

extern "C" void kernel_launch(void* const* d_in, const int* in_sizes, int n_in,
                              void* d_out, int out_size, void* d_ws, size_t ws_size,
                              hipStream_t stream) {
    const float* inputs = (const float*)d_in[0];
    const int*   t1_idx = (const int*)d_in[1];
    const float* t1_val = (const float*)d_in[2];
    const int*   t2_idx = (const int*)d_in[3];
    const float* t2_val = (const float*)d_in[4];
    const float* weight = (const float*)d_in[5];
    const float* bias   = (const float*)d_in[6];
    float* out = (float*)d_out;

    float* xs  = (float*)d_ws;               // 5 buffers of XSZ floats
    float* x0  = xs;
    float* x1a = xs + 1ull * XSZ;
    float* x2a = xs + 2ull * XSZ;
    float* x1b = xs + 3ull * XSZ;
    float* x2b = xs + 4ull * XSZ;

    const int INIT_BLKS = XSZ / 4 / 256;     // float4 per thread
    const int X0_BLKS   = XSZ / 256;

    // x0 = transpose(inputs)
    k_build_x0<<<X0_BLKS, 256, 0, stream>>>(inputs, x0);

    // Transition 1: x1 = A1 x0 ; x2 = 2 A1 x1 - x0
    k_init<<<INIT_BLKS, 256, 0, stream>>>(x0, x1a, 0.0f);
    k_spmm<<<E_EDGES, 256, 0, stream>>>(t1_idx, t1_idx + E_EDGES, t1_val, x0, x1a, 1.0f);
    k_init<<<INIT_BLKS, 256, 0, stream>>>(x0, x2a, -1.0f);
    k_spmm<<<E_EDGES, 256, 0, stream>>>(t1_idx, t1_idx + E_EDGES, t1_val, x1a, x2a, 2.0f);

    // Transition 2
    k_init<<<INIT_BLKS, 256, 0, stream>>>(x0, x1b, 0.0f);
    k_spmm<<<E_EDGES, 256, 0, stream>>>(t2_idx, t2_idx + E_EDGES, t2_val, x0, x1b, 1.0f);
    k_init<<<INIT_BLKS, 256, 0, stream>>>(x0, x2b, -1.0f);
    k_spmm<<<E_EDGES, 256, 0, stream>>>(t2_idx, t2_idx + E_EDGES, t2_val, x1b, x2b, 2.0f);

    // Fused gather + GEMM: 320000 rows / 64 per block, 64 threads (2 waves)
    k_gemm_wmma<<<(B_SZ * N_NODES) / 64, 64, 0, stream>>>(xs, weight, bias, out);
}